// LamdaLoss_11871289606443
// MI455X (gfx1250) — compile-verified
//
#include <hip/hip_runtime.h>
#include <hip/hip_bf16.h>

typedef __attribute__((ext_vector_type(2))) float v2f;
typedef __attribute__((ext_vector_type(8))) float v8f;

namespace {

constexpr int HW      = 512;
constexpr int CH      = 8;
constexpr int NPAIR   = 28;
constexpr int NBATCH  = 4;
constexpr int OUT_DIM = 450;                       // (512-31) valid conv, cropped [16:-15]
constexpr int TILE    = 32;
constexpr int NT      = (OUT_DIM + TILE - 1) / TILE; // 15
constexpr int XROWS   = 64;                        // 63 valid rows + 1 zero pad
constexpr int XS      = 65;                        // padded LDS col stride (odd -> fewer bank conflicts)
constexpr int TS      = 65;                        // T-scratch stride
constexpr int STS     = 33;                        // stats tile stride
constexpr int NWAVE   = 8;
constexpr float EPS   = 1e-8f;

constexpr int X_FLOATS   = CH * XROWS * XS;        // 33280
constexpr int T_FLOATS   = NWAVE * 32 * TS;        // 16640
constexpr int ST_FLOATS  = CH * 32 * STS;          // 8448
constexpr int LDS_FLOATS = X_FLOATS + T_FLOATS + 2 * ST_FLOATS; // 66816 f32 = 261 KB < 320 KB

__device__ __constant__ int PI_[NPAIR] = {0,0,0,0,0,0,0,1,1,1,1,1,1,2,2,2,2,2,3,3,3,3,4,4,4,5,5,6};
__device__ __constant__ int PJ_[NPAIR] = {1,2,3,4,5,6,7,2,3,4,5,6,7,3,4,5,6,7,4,5,6,7,5,6,7,6,7,7};

// ---------------------------------------------------------------------------
// 32x32 box-sum of one 63x63 LDS tile via banded GEMM:
//   T(32x64)  = Wv^T(32x64banded) * Bmap(64x64)      (vertical 32-tap sum)
//   Out(32x32)= T(32x64) * Wh(64x32banded)           (horizontal 32-tap sum)
// Band weights (0/1) are generated on the fly; Bmap element is x, x*x, or
// x_i*x_j depending on MODE. All WMMAs are V_WMMA_F32_16X16X4_F32.
// ---------------------------------------------------------------------------
template <int MODE>
__device__ __forceinline__ void box_tile(const float* __restrict__ Xa,
                                         const float* __restrict__ Xb,
                                         float* __restrict__ Tw,
                                         int lane, v8f acc[2][2]) {
  const int m  = lane & 15;       // row within A / col within B,C,D
  const int hi = lane >> 4;       // lane-half selects K pair / M+8
  const int kh = hi * 2;

  // ---- pass 1: vertical band sum, result staged to per-wave LDS scratch ----
  for (int mt = 0; mt < 2; ++mt) {
    const int t = mt * 16 + m;    // output row index (A-matrix M)
    for (int nt = 0; nt < 4; ++nt) {
      const int col = nt * 16 + m;
      v8f d = {};
      for (int kb = 4 * mt; kb < 4 * mt + 12; ++kb) {   // banded K range
        const int r = kb * 4 + kh;
        v2f a, b;
        a.x = (r     >= t && r     <= t + 31) ? 1.0f : 0.0f;  // Wv^T[t][r]
        a.y = (r + 1 >= t && r + 1 <= t + 31) ? 1.0f : 0.0f;
        float b0 = Xa[r * XS + col];
        float b1 = Xa[(r + 1) * XS + col];
        if (MODE == 1) { b0 *= b0; b1 *= b1; }
        if (MODE == 2) { b0 *= Xb[r * XS + col]; b1 *= Xb[(r + 1) * XS + col]; }
        b.x = b0; b.y = b1;
        d = __builtin_amdgcn_wmma_f32_16x16x4_f32(false, a, false, b,
                                                  (short)0, d, false, false);
      }
#pragma unroll
      for (int v = 0; v < 8; ++v)                       // D layout -> scratch
        Tw[(mt * 16 + v + 8 * hi) * TS + col] = d[v];
    }
  }
  // Same-wave DS ops are in-order; no barrier needed before re-reading Tw.

  // ---- pass 2: horizontal band sum, accumulators stay in VGPRs ----
  for (int mt = 0; mt < 2; ++mt) {
    for (int nt = 0; nt < 2; ++nt) {
      const int x = nt * 16 + m;  // output col (B-matrix N)
      v8f d = {};
      for (int kb = 4 * nt; kb < 4 * nt + 12; ++kb) {
        const int c0 = kb * 4 + kh;
        v2f a, b;
        a.x = Tw[(mt * 16 + m) * TS + c0];
        a.y = Tw[(mt * 16 + m) * TS + c0 + 1];
        b.x = (c0     >= x && c0     <= x + 31) ? 1.0f : 0.0f;  // Wh[c][x]
        b.y = (c0 + 1 >= x && c0 + 1 <= x + 31) ? 1.0f : 0.0f;
        d = __builtin_amdgcn_wmma_f32_16x16x4_f32(false, a, false, b,
                                                  (short)0, d, false, false);
      }
      acc[mt][nt] = d;
    }
  }
}

__device__ __forceinline__ void store_stats(float* __restrict__ dst,
                                            v8f acc[2][2], int lane) {
  const int m = lane & 15, hi = lane >> 4;
  for (int mt = 0; mt < 2; ++mt)
    for (int nt = 0; nt < 2; ++nt)
#pragma unroll
      for (int v = 0; v < 8; ++v)
        dst[(mt * 16 + v + 8 * hi) * STS + nt * 16 + m] = acc[mt][nt][v];
}

// ---------------------------------------------------------------------------
// Fused kernel: one block = one (input, batch, 32x32 output tile).
// Loads 8ch x 63x63 halo into LDS once, computes all box stats with WMMA,
// forms per-pixel Q for all 28 pairs, accumulates per-pair sums globally.
// ---------------------------------------------------------------------------
__global__ void lamda_fused_kernel(const float* __restrict__ pred,
                                   const float* __restrict__ lrms,
                                   float* __restrict__ qsum) {
  extern __shared__ float sm[];
  float* sm_X  = sm;                       // [8][64][XS]
  float* sm_T  = sm_X + X_FLOATS;          // [8 waves][32][TS]
  float* sm_MU = sm_T + T_FLOATS;          // [8][32][STS] raw box sums
  float* sm_M2 = sm_MU + ST_FLOATS;        // [8][32][STS]

  const int tileId = blockIdx.x;
  const int ty = tileId / NT, tx = tileId % NT;
  const int t0 = ty * TILE, x0 = tx * TILE;
  const int bb = blockIdx.y;
  const int zz = blockIdx.z;
  const float* __restrict__ src = (zz == 0) ? pred : lrms;

  // ---- cooperative halo load (zero-padded to 64x(XS)) ----
  for (int idx = threadIdx.x; idx < X_FLOATS; idx += blockDim.x) {
    const int c   = idx / (XROWS * XS);
    const int rem = idx - c * (XROWS * XS);
    const int r   = rem / XS;
    const int col = rem - r * XS;
    float v = 0.0f;
    if (r < 63 && col < 63) {
      const int rg = t0 + 16 + r;
      const int cg = x0 + 16 + col;
      if (rg < HW && cg < HW)
        v = src[((size_t)(bb * CH + c) * HW + rg) * HW + cg];
    }
    sm_X[idx] = v;
  }
  __syncthreads();

  const int wave = threadIdx.x >> 5;
  const int lane = threadIdx.x & 31;
  float* Tw = sm_T + wave * 32 * TS;
  v8f acc[2][2];

  // ---- phase 1: per-channel mu & m2 box sums (wave w owns channel w) ----
  {
    const float* Xc = sm_X + wave * XROWS * XS;
    box_tile<0>(Xc, Xc, Tw, lane, acc);
    store_stats(sm_MU + wave * 32 * STS, acc, lane);
    box_tile<1>(Xc, Xc, Tw, lane, acc);
    store_stats(sm_M2 + wave * 32 * STS, acc, lane);
  }
  __syncthreads();

  // ---- phase 2: per-pair mu12 + Q, waves stride over the 28 pairs ----
  const int m  = lane & 15;
  const int hi = lane >> 4;
  const float S = 1.0f / 1024.0f;          // box sum -> box mean (32*32)

  for (int p = wave; p < NPAIR; p += NWAVE) {
    const int ci = PI_[p], cj = PJ_[p];
    box_tile<2>(sm_X + ci * XROWS * XS, sm_X + cj * XROWS * XS, Tw, lane, acc);

    float qp = 0.0f;
    for (int mt = 0; mt < 2; ++mt)
      for (int nt = 0; nt < 2; ++nt)
#pragma unroll
        for (int v = 0; v < 8; ++v) {
          const int t = mt * 16 + v + 8 * hi;
          const int x = nt * 16 + m;
          if (t0 + t < OUT_DIM && x0 + x < OUT_DIM) {
            const float mu1  = sm_MU[ci * 32 * STS + t * STS + x] * S;
            const float mu2  = sm_MU[cj * 32 * STS + t * STS + x] * S;
            const float m2a  = sm_M2[ci * 32 * STS + t * STS + x] * S;
            const float m2b  = sm_M2[cj * 32 * STS + t * STS + x] * S;
            const float mu12 = acc[mt][nt][v] * S;
            const float sig1  = m2a - mu1 * mu1;
            const float sig2  = m2b - mu2 * mu2;
            const float sig12 = mu12 - mu1 * mu2;
            const float sv = sig1 + sig2;
            const float mv = mu1 * mu1 + mu2 * mu2;
            const float mm = 2.0f * mu1 * mu2;
            const float m_safe = (mv > EPS) ? mv : 1.0f;
            const float s_safe = (sv > EPS) ? sv : 1.0f;
            float q;
            if      (sv < EPS && mv > EPS) q = mm / m_safe;
            else if (sv > EPS && mv < EPS) q = 2.0f * sig12 / s_safe;
            else if (sv > EPS && mv > EPS) q = mm * (2.0f * sig12) / (m_safe * s_safe);
            else                           q = 1.0f;
            qp += q;
          }
        }
    // wave32 reduction, one atomic per (input, pair)
    for (int off = 16; off > 0; off >>= 1) qp += __shfl_xor(qp, off, 32);
    if (lane == 0) atomicAdd(&qsum[zz * NPAIR + p], qp);
  }
}

__global__ void zero_ws_kernel(float* ws) {
  if (threadIdx.x < 2 * NPAIR + 8) ws[threadIdx.x] = 0.0f;
}

__global__ void finalize_kernel(const float* __restrict__ qsum,
                                float* __restrict__ out) {
  const int t = threadIdx.x;
  const float invc = 1.0f / ((float)NBATCH * (float)OUT_DIM * (float)OUT_DIM);
  float v = 0.0f;
  if (t < NPAIR)
    v = fabsf(qsum[t] * invc - qsum[NPAIR + t] * invc);
  for (int off = 16; off > 0; off >>= 1) v += __shfl_down(v, off, 32);
  if (t == 0) out[0] = v * (1.0f / (float)NPAIR);
}

} // namespace

extern "C" void kernel_launch(void* const* d_in, const int* in_sizes, int n_in,
                              void* d_out, int out_size, void* d_ws, size_t ws_size,
                              hipStream_t stream) {
  (void)in_sizes; (void)n_in; (void)out_size; (void)ws_size;
  const float* pred = (const float*)d_in[0];
  const float* lrms = (const float*)d_in[1];
  float* qsum = (float*)d_ws;              // 56 accumulators
  float* out  = (float*)d_out;

  zero_ws_kernel<<<1, 64, 0, stream>>>(qsum);

  dim3 grid(NT * NT, NBATCH, 2);
  const size_t shbytes = (size_t)LDS_FLOATS * sizeof(float);
  lamda_fused_kernel<<<grid, 256, shbytes, stream>>>(pred, lrms, qsum);

  finalize_kernel<<<1, 32, 0, stream>>>(qsum, out);
}